// OnlineAuSPRO_80290118632120
// MI455X (gfx1250) — compile-verified
//
#include <hip/hip_runtime.h>
#include <hip/hip_bf16.h>

typedef __attribute__((ext_vector_type(2))) float v2f;
typedef __attribute__((ext_vector_type(8))) float v8f;

// Problem constants (match reference)
#define Bn    64
#define Hn    512
#define Wn    512
#define HWn   (Hn * Wn)          // 262144 pixels per image
#define Tn    100                // thresholds
#define TP1   101                // buckets 0..T
#define HPAD  104                // padded bins per segment (mult of 4 for WMMA K)
#define Rn    8                  // regions per image
#define NSEG  (Bn * Rn)          // 512 segments
#define CHUNKS 8                 // blocks per image in histogram pass
#define VEC_PER_BLK ((HWn / CHUNKS) / 4)  // 8192 float4/int4 per block
#define SATF  1.0f

// Workspace layout in 32-bit units
#define OFF_BG    0
#define OFF_HIST  (OFF_BG + HPAD)                 // [NSEG][HPAD] ints
#define OFF_AREA  (OFF_HIST + NSEG * HPAD)       // [NSEG] floats
#define OFF_PART  (OFF_AREA + NSEG)              // [112][NM_TILES] floats
#define NT_TILES  7                               // ceil(100/16)
#define NM_TILES  (NSEG / 16)                     // 32
#define PART_SZ   (NT_TILES * 16 * NM_TILES)      // 112*32
#define WS_INTS   (OFF_PART + PART_SZ)

// ---------------------------------------------------------------- zero ws
__global__ void auspro_zero_ws(int* ws) {
    int i = blockIdx.x * blockDim.x + threadIdx.x;
    if (i < WS_INTS) ws[i] = 0;
}

// ------------------------------------------------- searchsorted (right)
__device__ __forceinline__ int bucket_of(float s, const float* sthr) {
    int lo = 0, hi = Tn;                 // first idx with thr[idx] > s
    while (lo < hi) {
        int mid = (lo + hi) >> 1;
        if (sthr[mid] <= s) lo = mid + 1; else hi = mid;
    }
    return lo;                           // in [0, Tn]
}

// ----------------------------------------------- K1: streaming histogram
// grid: (CHUNKS, Bn), block: 256. LDS-privatized bg + per-region hists.
__global__ void auspro_hist(const float* __restrict__ preds,
                            const float* __restrict__ thr,
                            const int*   __restrict__ labels,
                            int* __restrict__ g_bg,      // [HPAD]
                            int* __restrict__ g_hist) {  // [NSEG][HPAD]
    __shared__ float s_thr[Tn];
    __shared__ int   s_bg[HPAD];
    __shared__ int   s_reg[Rn][HPAD];
    const int tid = threadIdx.x;
    for (int i = tid; i < Tn; i += 256) s_thr[i] = thr[i];
    for (int i = tid; i < HPAD; i += 256) s_bg[i] = 0;
    for (int i = tid; i < Rn * HPAD; i += 256) ((int*)s_reg)[i] = 0;
    __syncthreads();

    const int b = blockIdx.y;
    const long base = (long)b * HWn + (long)blockIdx.x * (HWn / CHUNKS);
    const float4* p4 = (const float4*)(preds  + base);
    const int4*   l4 = (const int4*)  (labels + base);

    for (int i = tid; i < VEC_PER_BLK; i += 256) {
        if (i + 256 < VEC_PER_BLK) {            // gfx1250 global_prefetch_b8
            __builtin_prefetch(&p4[i + 256], 0, 0);
            __builtin_prefetch(&l4[i + 256], 0, 0);
        }
        float4 s = p4[i];
        int4   l = l4[i];
        const float sv[4] = {s.x, s.y, s.z, s.w};
        const int   lv[4] = {l.x, l.y, l.z, l.w};
#pragma unroll
        for (int c = 0; c < 4; ++c) {
            int bk = bucket_of(sv[c], s_thr);
            int lab = lv[c];
            if (lab == 0)                 atomicAdd(&s_bg[bk], 1);
            else if (lab >= 1 && lab <= Rn) atomicAdd(&s_reg[lab - 1][bk], 1);
        }
    }
    __syncthreads();

    for (int i = tid; i < HPAD; i += 256) {
        int v = s_bg[i];
        if (v) atomicAdd(&g_bg[i], v);
    }
    for (int i = tid; i < Rn * HPAD; i += 256) {
        int v = ((int*)s_reg)[i];
        if (v) {
            int r = i / HPAD, k = i - r * HPAD;
            atomicAdd(&g_hist[(b * Rn + r) * HPAD + k], v);
        }
    }
}

// ------------------------------------------------------- K2: region areas
__global__ void auspro_area(const int* __restrict__ g_hist, float* __restrict__ area) {
    int s = blockIdx.x * blockDim.x + threadIdx.x;
    if (s < NSEG) {
        int a = 0;
        for (int k = 0; k < TP1; ++k) a += g_hist[s * HPAD + k];
        area[s] = (float)a;
    }
}

// ------------------------ K3: suffix sums TP via f32 WMMA (exact: ints<2^24)
// TP[seg][t] = sum_{k>t} hist[seg][k].  D(16seg x 16t) += A(16x4 hist) * B(4x16 tri)
// A layout (ISA 7.12.2, 32-bit 16x4): lane&15 = M(seg); lanes<16 K={k0,k0+1},
// lanes>=16 K={k0+2,k0+3}. B built in-register mirroring A (N=lane&15, same K split).
// C/D layout: VGPR j -> M=j (lanes<16) / M=8+j (lanes>=16), N=lane&15.
__global__ void auspro_spro_wmma(const int*  __restrict__ g_hist,   // [NSEG][HPAD]
                                 const float* __restrict__ area,     // [NSEG]
                                 float* __restrict__ part) {         // [112][NM_TILES]
    __shared__ float s_red[32];
    const int lane   = threadIdx.x;          // 32 threads = 1 wave
    const int Mbase  = blockIdx.x * 16;      // segment tile
    const int tbase  = blockIdx.y * 16;      // threshold tile
    const bool hiH   = lane >= 16;
    const int  seg   = Mbase + (lane & 15);
    const int  t     = tbase + (lane & 15);

    v8f c = {};
    for (int k0 = 0; k0 < HPAD; k0 += 4) {
        const int kk = k0 + (hiH ? 2 : 0);
        v2f a, bv;
        a.x = (float)g_hist[seg * HPAD + kk];
        a.y = (float)g_hist[seg * HPAD + kk + 1];
        bv.x = (kk     > t) ? 1.0f : 0.0f;   // strict upper-triangular ones
        bv.y = (kk + 1 > t) ? 1.0f : 0.0f;
        c = __builtin_amdgcn_wmma_f32_16x16x4_f32(false, a, false, bv,
                                                  (short)0, c, false, false);
    }

    // sPRO contribution of this wave's 16 segments at threshold t
    float p = 0.0f;
    if (t < Tn) {
#pragma unroll
        for (int j = 0; j < 8; ++j) {
            int sg = Mbase + j + (hiH ? 8 : 0);
            float ar = area[sg];
            if (ar > 0.0f) {
                float sat = fmaxf(ar * SATF, 1.0f);
                p += fminf(c[j] / sat, 1.0f);
            }
        }
    }
    s_red[lane] = p;
    __syncthreads();
    if (lane < 16) {                         // combine the two M-halves
        float v = s_red[lane] + s_red[lane + 16];
        part[(blockIdx.y * 16 + lane) * NM_TILES + blockIdx.x] = v;
    }
}

// --------------------------------------------- K4: fpr, sort, trapezoid AUC
__global__ void auspro_finalize(const int*  __restrict__ g_bg,
                                const float* __restrict__ area,
                                const float* __restrict__ part,
                                float* __restrict__ out) {
    __shared__ float red[128];
    __shared__ int   cum[TP1];
    __shared__ int   s_total;
    __shared__ float s_ndef;
    __shared__ float fpr[Tn], msp[Tn], xs[Tn], ys[Tn];
    const int tid = threadIdx.x;             // 128 threads

    // n_def = max(#regions with area>0, 1)
    float v = 0.0f;
    for (int i = tid; i < NSEG; i += 128) v += (area[i] > 0.0f) ? 1.0f : 0.0f;
    red[tid] = v; __syncthreads();
    for (int s = 64; s > 0; s >>= 1) { if (tid < s) red[tid] += red[tid + s]; __syncthreads(); }
    if (tid == 0) {
        s_ndef = fmaxf(red[0], 1.0f);
        int acc = 0;
        for (int k = 0; k < TP1; ++k) { acc += g_bg[k]; cum[k] = acc; }
        s_total = acc;
    }
    __syncthreads();

    if (tid < Tn) {
        float fp = (float)(s_total - cum[tid]);        // #bg with score >= thr[t]
        float tn = (float)s_total - fp;
        float denom = fp + tn;
        fpr[tid] = (denom > 0.0f) ? fp / fmaxf(denom, 1.0f) : 0.0f;
        float ss = 0.0f;
        for (int m = 0; m < NM_TILES; ++m) ss += part[tid * NM_TILES + m];  // fixed order
        msp[tid] = ss / s_ndef;
    }
    __syncthreads();

    if (tid < Tn) {                          // stable argsort via rank counting
        float xi = fpr[tid]; int rank = 0;
        for (int j = 0; j < Tn; ++j) {
            float xj = fpr[j];
            rank += (xj < xi) || (xj == xi && j < tid);
        }
        xs[rank] = xi; ys[rank] = msp[tid];
    }
    __syncthreads();

    if (tid == 0) {
        float auc = 0.0f;
        for (int i = 0; i + 1 < Tn; ++i)
            auc += (xs[i + 1] - xs[i]) * (ys[i + 1] + ys[i]) * 0.5f;
        out[0] = auc;
    }
}

// ---------------------------------------------------------------- launch
extern "C" void kernel_launch(void* const* d_in, const int* in_sizes, int n_in,
                              void* d_out, int out_size, void* d_ws, size_t ws_size,
                              hipStream_t stream) {
    const float* preds  = (const float*)d_in[0];
    const float* thr    = (const float*)d_in[1];
    const int*   labels = (const int*)  d_in[2];
    int*   ws  = (int*)d_ws;
    float* out = (float*)d_out;

    auspro_zero_ws<<<dim3((WS_INTS + 255) / 256), dim3(256), 0, stream>>>(ws);
    auspro_hist<<<dim3(CHUNKS, Bn), dim3(256), 0, stream>>>(
        preds, thr, labels, ws + OFF_BG, ws + OFF_HIST);
    auspro_area<<<dim3((NSEG + 255) / 256), dim3(256), 0, stream>>>(
        ws + OFF_HIST, (float*)(ws + OFF_AREA));
    auspro_spro_wmma<<<dim3(NM_TILES, NT_TILES), dim3(32), 0, stream>>>(
        ws + OFF_HIST, (const float*)(ws + OFF_AREA), (float*)(ws + OFF_PART));
    auspro_finalize<<<dim3(1), dim3(128), 0, stream>>>(
        ws + OFF_BG, (const float*)(ws + OFF_AREA), (const float*)(ws + OFF_PART), out);
}